// PatchApplier_55121610277482
// MI455X (gfx1250) — compile-verified
//
#include <hip/hip_runtime.h>
#include <cstdint>

// ---------------------------------------------------------------------------
// PatchApplier for MI455X (gfx1250).
// Memory-bound op: ~650MB of HBM traffic, ~0.2 GFLOP of math -> pure streaming.
// Kernel 1: NT float4 copy images->out (keeps 196KB patch L2-resident).
// Kernel 2: per-(image,channel) block stages patch channel into LDS with
//           gfx1250 async global->LDS b128 ops (ASYNCcnt), then bilinear
//           samples the <=153x153 paste rectangle and overwrites out.
// ---------------------------------------------------------------------------

typedef __attribute__((ext_vector_type(4))) float v4f;
typedef __attribute__((ext_vector_type(4))) int   v4i;

#define AS1 __attribute__((address_space(1)))
#define AS3 __attribute__((address_space(3)))

typedef AS1 v4i* glb_v4i_ptr;
typedef AS3 v4i* lds_v4i_ptr;

#define IMG_B 64
#define IMG_C 3
#define IMG_H 640
#define IMG_W 640
#define P_H 128
#define P_W 128

// ---- async global -> LDS staging (gfx1250 path with safe fallback) --------
__device__ __forceinline__ void async_copy_f4_to_lds(const float* g, float* l) {
#if __has_builtin(__builtin_amdgcn_global_load_async_to_lds_b128)
  // per-lane: copies 16 bytes from global address g to LDS address l, tracked by ASYNCcnt
  __builtin_amdgcn_global_load_async_to_lds_b128(
      (glb_v4i_ptr)(uintptr_t)g,
      (lds_v4i_ptr)(uint32_t)(uintptr_t)l,
      /*imm offset*/ 0, /*cpol*/ 0);
#else
  *(v4f*)l = *(const v4f*)g;
#endif
}

__device__ __forceinline__ void wait_async_lds() {
#if __has_builtin(__builtin_amdgcn_global_load_async_to_lds_b128)
# if __has_builtin(__builtin_amdgcn_s_wait_asynccnt)
  __builtin_amdgcn_s_wait_asynccnt(0);
# else
  asm volatile("s_wait_asynccnt 0" ::: "memory");
# endif
#endif
}

// ---- Kernel 1: bulk NT copy (float4 per thread) ----------------------------
__global__ __launch_bounds__(256) void pa_copy_kernel(const v4f* __restrict__ in,
                                                      v4f* __restrict__ out,
                                                      int n4) {
  int i = blockIdx.x * 256 + threadIdx.x;
  if (i < n4) {
    v4f v = __builtin_nontemporal_load(&in[i]);
    __builtin_nontemporal_store(v, &out[i]);
  }
}

// ---- Kernel 2: paste rectangle, one block per (image, channel) -------------
__global__ __launch_bounds__(256) void pa_paste_kernel(const float* __restrict__ patch,
                                                       const float* __restrict__ angles,
                                                       const float* __restrict__ scales,
                                                       const float* __restrict__ ux,
                                                       const float* __restrict__ uy,
                                                       float* __restrict__ out) {
  __shared__ float sp[P_H * P_W];  // 64 KB: one patch channel

  const int b = blockIdx.x;
  const int c = blockIdx.y;
  const int tid = threadIdx.x;

  // Stage patch channel c into LDS: 4096 float4s, 16 async b128 ops per lane.
  const float* pc = patch + (size_t)c * (P_H * P_W);
  #pragma unroll
  for (int k = 0; k < (P_H * P_W) / (4 * 256); ++k) {
    int i4 = (tid + k * 256) * 4;
    async_copy_f4_to_lds(pc + i4, sp + i4);
  }
  wait_async_lds();
  __syncthreads();

  // Per-image parameters (mirror reference f32 math exactly).
  const float s   = scales[b];
  const float shf = floorf((float)P_H * s);
  const float swf = floorf((float)P_W * s);
  const float y0f = floorf(uy[b] * ((float)IMG_H - shf));
  const float x0f = floorf(ux[b] * ((float)IMG_W - swf));
  const float theta = angles[b] * 0.017453292519943295f;  // deg2rad
  float si, co;
  sincosf(theta, &si, &co);
  const float inv_sy = (float)P_H / shf;
  const float inv_sx = (float)P_W / swf;
  const float cy = ((float)P_H - 1.0f) * 0.5f;  // 63.5
  const float cx = ((float)P_W - 1.0f) * 0.5f;

  const int shi = (int)shf;
  const int swi = (int)swf;
  const int y0i = (int)y0f;
  const int x0i = (int)x0f;
  const int npix = shi * swi;

  float* outc = out + (((size_t)b * IMG_C + c) * IMG_H) * IMG_W;

  for (int p = tid; p < npix; p += 256) {
    const int ry = p / swi;            // yl
    const int rx = p - ry * swi;       // xl
    // inverse resize
    const float yr = ((float)ry + 0.5f) * inv_sy - 0.5f;
    const float xr = ((float)rx + 0.5f) * inv_sx - 0.5f;
    // inverse rotation about patch center
    const float dy = yr - cy;
    const float dx = xr - cx;
    const float xp = co * dx + si * dy + cx;
    const float yp = -si * dx + co * dy + cy;

    const float yfl = floorf(yp);
    const float xfl = floorf(xp);
    const float wy = yp - yfl;
    const float wx = xp - xfl;
    const int yi = (int)yfl;
    const int xi = (int)xfl;

    // zero-padded gather from LDS-resident patch channel
    auto g = [&](int yy, int xx) -> float {
      const bool inb = (yy >= 0) & (yy <= P_H - 1) & (xx >= 0) & (xx <= P_W - 1);
      const int yc = min(max(yy, 0), P_H - 1);
      const int xc = min(max(xx, 0), P_W - 1);
      return inb ? sp[yc * P_W + xc] : 0.0f;
    };

    const float v = g(yi, xi)         * (1.0f - wy) * (1.0f - wx)
                  + g(yi, xi + 1)     * (1.0f - wy) * wx
                  + g(yi + 1, xi)     * wy          * (1.0f - wx)
                  + g(yi + 1, xi + 1) * wy          * wx;

    outc[(size_t)(y0i + ry) * IMG_W + (x0i + rx)] = v;
  }
}

// ---------------------------------------------------------------------------
extern "C" void kernel_launch(void* const* d_in, const int* in_sizes, int n_in,
                              void* d_out, int out_size, void* d_ws, size_t ws_size,
                              hipStream_t stream) {
  (void)in_sizes; (void)n_in; (void)out_size; (void)d_ws; (void)ws_size;

  const float* images = (const float*)d_in[0];
  const float* patch  = (const float*)d_in[1];
  const float* angles = (const float*)d_in[2];
  const float* scales = (const float*)d_in[3];
  const float* ux     = (const float*)d_in[4];
  const float* uy     = (const float*)d_in[5];
  float* out = (float*)d_out;

  const int n4 = (IMG_B * IMG_C * IMG_H * IMG_W) / 4;  // 19,660,800 float4s
  pa_copy_kernel<<<n4 / 256, 256, 0, stream>>>((const v4f*)images, (v4f*)out, n4);

  pa_paste_kernel<<<dim3(IMG_B, IMG_C), 256, 0, stream>>>(
      patch, angles, scales, ux, uy, out);
}